// rf_44186623541739
// MI455X (gfx1250) — compile-verified
//
#include <hip/hip_runtime.h>
#include <math.h>

// Problem constants from the reference
#define BS  512
#define SEQ 128
#define FF  136
#define HH  128

typedef float v2f __attribute__((ext_vector_type(2)));
typedef float v8f __attribute__((ext_vector_type(8)));

__device__ __forceinline__ float sigmoidf_(float x) { return 1.0f / (1.0f + __expf(-x)); }

// ---------------------------------------------------------------------------
// f32 GEMM with V_WMMA_F32_16X16X4_F32.
// out[M,N] = act(A[M,K] @ W[K,N] + bias),  act: 0=none, 1=relu, 2=sigmoid.
// 256 threads = 8 wave32s; wave w computes the 16x16 tile at column
// n0 = 16*(8*blockIdx.y + w).  A-tile (16x4 f32): lanes 0-15 carry K=0,1 in
// vgpr0/1; lanes 16-31 carry K=2,3.  B-tile (4x16) mirrors with N striped
// across lanes.  C/D: vgpr r holds M=r (lanes 0-15) / M=r+8 (lanes 16-31).
// tr_out=1 stores transposed per 128-row batch block:
//   out[(m>>7)*128*128 + n*128 + (m&127)]   (pays the scatter once so the
//   sequential kernel's 128 per-step reads are coalesced).
// ---------------------------------------------------------------------------
__global__ void __launch_bounds__(256) wmma_gemm_f32(
    const float* __restrict__ A, const float* __restrict__ W,
    const float* __restrict__ bias, float* __restrict__ out,
    int M, int K, int N, int act, int tr_out)
{
  const int tid  = threadIdx.x;
  const int wave = tid >> 5;
  const int lane = tid & 31;
  const int half = lane >> 4;
  const int l    = lane & 15;
  const int m0   = blockIdx.x * 16;
  const int n0   = (blockIdx.y * 8 + wave) * 16;
  if (n0 >= N) return;

  v8f c = {};
  const float* arow = A + (size_t)(m0 + l) * K + 2 * half;
  const float* wcol = W + (size_t)(2 * half) * N + (n0 + l);
  for (int k = 0; k < K; k += 4) {
    v2f a, b;
    a.x = arow[k];
    a.y = arow[k + 1];
    b.x = wcol[(size_t)k * N];
    b.y = wcol[(size_t)(k + 1) * N];
    c = __builtin_amdgcn_wmma_f32_16x16x4_f32(false, a, false, b, (short)0, c,
                                              false, false);
  }

  const float bv = bias ? bias[n0 + l] : 0.0f;
#pragma unroll
  for (int r = 0; r < 8; ++r) {
    float v = c[r] + bv;
    if (act == 1)      v = fmaxf(v, 0.0f);
    else if (act == 2) v = sigmoidf_(v);
    const int m = m0 + r + 8 * half;
    size_t o = tr_out ? (((size_t)(m >> 7)) << 14) + (size_t)(n0 + l) * 128 + (m & 127)
                      : (size_t)m * N + (n0 + l);
    out[o] = v;
  }
}

// Simple transpose: in[R,C] -> out[C,R] (GRU weights, once).
__global__ void transpose_kernel(const float* __restrict__ in,
                                 float* __restrict__ out, int R, int C)
{
  int i = blockIdx.x * blockDim.x + threadIdx.x;
  if (i < R * C) {
    int r = i / C, c = i % C;
    out[(size_t)c * R + r] = in[i];
  }
}

// ---------------------------------------------------------------------------
// Sequential selection: one block per batch item (512 blocks x 8 wave32s ->
// full WGP occupancy), looping all 128 steps.  Hot data (docWT/docGI/weights)
// is L2-resident; per-trajectory state lives in ~9 KB of LDS.
// Only the per-step argmax index matters: ranking[b, idx_i] = 0.9^i.
// ---------------------------------------------------------------------------
__global__ void __launch_bounds__(256) seq_select_kernel(
    const float* __restrict__ batch_ranks,  // BS x SEQ x FF
    const float* __restrict__ doc,          // BS x SEQ x HH
    const float* __restrict__ docWT,        // BS x HH x SEQ (transposed doc@W1a + com_b1)
    const float* __restrict__ docGI,        // BS x SEQ x 3H = doc@Wih^T + bih, or null
    const float* __restrict__ cnt_W1, const float* __restrict__ cnt_b1,
    const float* __restrict__ cnt_W2, const float* __restrict__ cnt_b2,
    const float* __restrict__ com_W1,       // 3H x H (rows H..3H used here)
    const float* __restrict__ com_W2,       // H x 1
    const float* __restrict__ WihT,         // H x 3H (fallback when docGI==null)
    const float* __restrict__ WhhT,         // H x 3H
    const float* __restrict__ bih, const float* __restrict__ bhh,
    float* __restrict__ ranking)            // BS x SEQ
{
  __shared__ float ssum[FF];        // running masked sum per feature
  __shared__ float cat[2 * FF];     // [mx | mn]
  __shared__ int   smask[SEQ];
  __shared__ float sh[HH];          // GRU hidden
  __shared__ float z1[HH];
  __shared__ float cr[HH];
  __shared__ float hvec[HH];        // cr@W1b + h@W1c
  __shared__ float scores[SEQ];
  __shared__ float xsel[HH];
  __shared__ float sgi[3 * HH];
  __shared__ float sgh[3 * HH];
  __shared__ float w2[HH];
  __shared__ float scnt, spw;
  __shared__ int   sidx;

  const int b   = blockIdx.x;
  const int tid = threadIdx.x;
  const float* br  = batch_ranks + (size_t)b * SEQ * FF;
  const float* dwt = docWT + ((size_t)b << 14);

  if (tid < SEQ) smask[tid] = 1;
  if (tid < HH)  { sh[tid] = 0.0f; w2[tid] = com_W2[tid]; }
  for (int f = tid; f < FF; f += 256) {
    float s = 0.0f;
    for (int q = 0; q < SEQ; ++q) s += br[(size_t)q * FF + f];
    ssum[f] = s;
  }
  if (tid == 0) { scnt = (float)SEQ; spw = 1.0f; }
  __syncthreads();

  for (int step = 0; step < SEQ; ++step) {
    // masked max + running mean per feature (coalesced over f, L2-resident)
    for (int f = tid; f < FF; f += 256) {
      float m = -INFINITY;
      for (int q = 0; q < SEQ; ++q)
        if (smask[q]) m = fmaxf(m, br[(size_t)q * FF + f]);
      cat[f]      = m;
      cat[FF + f] = ssum[f] / scnt;
    }
    __syncthreads();

    // cr = sigmoid(relu([mx,mn]@cnt_W1 + b1)@cnt_W2 + b2)
    if (tid < HH) {
      float acc = cnt_b1[tid];
      for (int f = 0; f < 2 * FF; ++f) acc += cat[f] * cnt_W1[(size_t)f * HH + tid];
      z1[tid] = fmaxf(acc, 0.0f);
    }
    __syncthreads();
    if (tid < HH) {
      float acc = cnt_b2[tid];
      for (int k = 0; k < HH; ++k) acc += z1[k] * cnt_W2[(size_t)k * HH + tid];
      cr[tid] = sigmoidf_(acc);
    }
    __syncthreads();

    // hvec = cr @ com_W1[H:2H] + h @ com_W1[2H:3H]   (doc part precomputed)
    if (tid < HH) {
      float acc = 0.0f;
      for (int k = 0; k < HH; ++k)
        acc += cr[k] * com_W1[(size_t)(HH + k) * HH + tid]
             + sh[k] * com_W1[(size_t)(2 * HH + k) * HH + tid];
      hvec[tid] = acc;
    }
    __syncthreads();

    // scores[s] = relu(docWT[b,:,s] + hvec) . com_W2  — coalesced over s
    if (tid < SEQ) {
      float acc = 0.0f;
      for (int k = 0; k < HH; ++k)
        acc += fmaxf(dwt[(size_t)k * SEQ + tid] + hvec[k], 0.0f) * w2[k];
      scores[tid] = acc;
    }
    __syncthreads();

    // masked argmax (first max wins, matching jnp.argmax)
    if (tid == 0) {
      float best = -INFINITY; int bi = 0;
      for (int q = 0; q < SEQ; ++q)
        if (smask[q] && scores[q] > best) { best = scores[q]; bi = q; }
      sidx = bi;
      smask[bi] = 0;
      ranking[(size_t)b * SEQ + bi] = spw;  // 0.9^step via inverse permutation
      spw  *= 0.9f;
      scnt -= 1.0f;
    }
    __syncthreads();

    const int idx = sidx;
    // Pull the selected rows toward the WGP early (global_prefetch_b8).
    if (docGI && tid == 0)
      __builtin_prefetch(docGI + ((size_t)b * SEQ + idx) * (3 * HH), 0, 1);
    for (int f = tid; f < FF; f += 256) ssum[f] -= br[(size_t)idx * FF + f];
    if (!docGI && tid < HH) xsel[tid] = doc[((size_t)b * SEQ + idx) * HH + tid];
    __syncthreads();

    // GRU gates: gi from precomputed docGI row (lookup) or GEMV fallback;
    // gh = h @ Whh^T + bhh (coalesced via WhhT).
    if (docGI) {
      const float* gir = docGI + ((size_t)b * SEQ + idx) * (3 * HH);
      for (int j = tid; j < 3 * HH; j += 256) {
        float gh = bhh[j];
        for (int k = 0; k < HH; ++k) gh += sh[k] * WhhT[(size_t)k * (3 * HH) + j];
        sgi[j] = gir[j]; sgh[j] = gh;
      }
    } else {
      for (int j = tid; j < 3 * HH; j += 256) {
        float gi = bih[j], gh = bhh[j];
        for (int k = 0; k < HH; ++k) {
          gi += xsel[k] * WihT[(size_t)k * (3 * HH) + j];
          gh += sh[k]   * WhhT[(size_t)k * (3 * HH) + j];
        }
        sgi[j] = gi; sgh[j] = gh;
      }
    }
    __syncthreads();
    if (tid < HH) {
      float r = sigmoidf_(sgi[tid] + sgh[tid]);
      float z = sigmoidf_(sgi[HH + tid] + sgh[HH + tid]);
      float n = tanhf(sgi[2 * HH + tid] + r * sgh[2 * HH + tid]);
      sh[tid] = (1.0f - z) * n + z * sh[tid];
    }
    __syncthreads();
  }
}

extern "C" void kernel_launch(void* const* d_in, const int* in_sizes, int n_in,
                              void* d_out, int out_size, void* d_ws, size_t ws_size,
                              hipStream_t stream)
{
  const float* batch_ranks = (const float*)d_in[0];
  // d_in[1] (label) only feeds the discarded losses — unused.
  const float* fnn_W1 = (const float*)d_in[2];
  const float* fnn_b1 = (const float*)d_in[3];
  const float* fnn_W2 = (const float*)d_in[4];
  const float* fnn_b2 = (const float*)d_in[5];
  const float* cnt_W1 = (const float*)d_in[6];
  const float* cnt_b1 = (const float*)d_in[7];
  const float* cnt_W2 = (const float*)d_in[8];
  const float* cnt_b2 = (const float*)d_in[9];
  const float* com_W1 = (const float*)d_in[10];
  const float* com_b1 = (const float*)d_in[11];
  const float* com_W2 = (const float*)d_in[12];
  const float* gru_Wih = (const float*)d_in[14];
  const float* gru_Whh = (const float*)d_in[15];
  const float* gru_bih = (const float*)d_in[16];
  const float* gru_bhh = (const float*)d_in[17];

  float* ws = (float*)d_ws;
  const size_t per = (size_t)BS * SEQ * HH;        // 8.39M floats per tensor
  const size_t wT  = (size_t)3 * HH * HH;          // 49152 floats
  float* doc   = ws;                                // sigmoid FNN output
  float* docWT = ws + per;                          // hidden temp, then transposed doc@W1a
  float* WihT  = ws + 2 * per;
  float* WhhT  = WihT + wT;
  float* docGI = WhhT + wT;                         // optional, 25.17M floats

  const size_t base_f   = 2 * per + 2 * wT;
  const size_t withGI_f = base_f + (size_t)BS * SEQ * 3 * HH;
  const bool useGI = ws_size >= withGI_f * sizeof(float);

  const int M = BS * SEQ;                           // 65536 rows

  transpose_kernel<<<(3 * HH * HH + 255) / 256, 256, 0, stream>>>(gru_Wih, WihT, 3 * HH, HH);
  transpose_kernel<<<(3 * HH * HH + 255) / 256, 256, 0, stream>>>(gru_Whh, WhhT, 3 * HH, HH);

  // hidden = relu(batch_ranks @ fnn_W1 + b1)           (into docWT buffer, plain)
  wmma_gemm_f32<<<dim3(M / 16, 1), 256, 0, stream>>>(batch_ranks, fnn_W1, fnn_b1,
                                                     docWT, M, FF, HH, 1, 0);
  // doc = sigmoid(hidden @ fnn_W2 + b2)
  wmma_gemm_f32<<<dim3(M / 16, 1), 256, 0, stream>>>(docWT, fnn_W2, fnn_b2,
                                                     doc, M, HH, HH, 2, 0);
  // docWT = transpose_per_batch(doc @ com_W1[0:128,:] + com_b1)
  wmma_gemm_f32<<<dim3(M / 16, 1), 256, 0, stream>>>(doc, com_W1, com_b1,
                                                     docWT, M, HH, HH, 0, 1);
  // docGI = doc @ Wih^T + bih   (hoists the GRU input-gate GEMM out of the loop)
  if (useGI)
    wmma_gemm_f32<<<dim3(M / 16, 3), 256, 0, stream>>>(doc, WihT, gru_bih,
                                                       docGI, M, HH, 3 * HH, 0, 0);

  seq_select_kernel<<<BS, 256, 0, stream>>>(batch_ranks, doc, docWT,
      useGI ? docGI : (const float*)nullptr,
      cnt_W1, cnt_b1, cnt_W2, cnt_b2, com_W1, com_W2, WihT, WhhT,
      gru_bih, gru_bhh, (float*)d_out);

  (void)in_sizes; (void)n_in; (void)out_size;
}